// TinyAttention_55576876810795
// MI455X (gfx1250) — compile-verified
//
#include <hip/hip_runtime.h>
#include <math.h>

// ---------------------------------------------------------------------------
// TinyAttention for MI455X (gfx1250, wave32):
//  - native fp32 WMMA (v_wmma_f32_16x16x4_f32) for all matmul stages
//  - ALL matrix tiles staged by the Tensor Data Mover (tensor_load_to_lds)
//    with hardware row padding, double-buffered with s_wait_tensorcnt(2)
//  - weights pre-transposed once (and V stored d-major) so every operand is
//    a plain 2D strided tile -> zero per-thread staging in the hot loops
// ---------------------------------------------------------------------------

typedef float v2f __attribute__((ext_vector_type(2)));
typedef float v8f __attribute__((ext_vector_type(8)));
typedef unsigned int u32x4 __attribute__((ext_vector_type(4)));
typedef int i32x4 __attribute__((ext_vector_type(4)));
typedef int i32x8 __attribute__((ext_vector_type(8)));

#define B_  2
#define S_  2048
#define H_  1024
#define NH_ 16
#define HD_ 64
#define EPS_ 1e-5f

#if defined(__has_builtin)
#  if __has_builtin(__builtin_amdgcn_tensor_load_to_lds) && \
      __has_builtin(__builtin_amdgcn_s_wait_tensorcnt)
#    define HAVE_TDM 1
#  endif
#endif
#ifndef HAVE_TDM
#  define HAVE_TDM 0
#endif

__device__ __forceinline__ v8f wmma_k4(v2f a, v2f b, v8f c) {
  return __builtin_amdgcn_wmma_f32_16x16x4_f32(
      /*neg_a=*/false, a, /*neg_b=*/false, b,
      /*c_mod=*/(short)0, c, /*reuse_a=*/false, /*reuse_b=*/false);
}

#if HAVE_TDM
// 2D tile load global->LDS via TDM (D# per CDNA5 ISA).
// tile_w 4B-elems per row, tile_h rows, row stride in elems.
// LDS row padding: pad_interval_code c -> pad every 2^(c+1) dwords,
// pad_amount_code a -> insert (a+1) dwords.
__device__ __forceinline__ void tdm_load_2d(const void* gsrc, unsigned lds_off,
                                            unsigned tile_w, unsigned tile_h,
                                            unsigned stride_elems,
                                            unsigned pad_interval_code,
                                            unsigned pad_amount_code,
                                            unsigned pad_enable) {
  unsigned long long ga = (unsigned long long)(uintptr_t)gsrc;
  u32x4 g0 = {};
  g0[0] = 1u;                                      // count=1 (user, valid)
  g0[1] = lds_off;                                 // lds_addr (bytes)
  g0[2] = (unsigned)ga;                            // global_addr[31:0]
  g0[3] = (unsigned)((ga >> 32) & 0x01ffffffull)   // global_addr[56:32]
          | (2u << 30);                            // type=2 ("image")
  i32x8 g1 = {};
  g1[0] = (int)((2u << 16)                         // data_size=2 -> 4 bytes
                | (pad_enable << 20)
                | (pad_interval_code << 22)
                | (pad_amount_code << 25));
  g1[1] = (int)((tile_w & 0xffffu) << 16);         // tensor_dim0[15:0]
  g1[2] = (int)(((tile_w >> 16) & 0xffffu)         // tensor_dim0[31:16]
                | ((tile_h & 0xffffu) << 16));     // tensor_dim1[15:0]
  g1[3] = (int)(((tile_h >> 16) & 0xffffu)         // tensor_dim1[31:16]
                | ((tile_w & 0xffffu) << 16));     // tile_dim0
  g1[4] = (int)(tile_h & 0xffffu);                 // tile_dim1 (tile_dim2=0)
  g1[5] = (int)stride_elems;                       // tensor_dim0_stride[31:0]
  g1[6] = 0;
  g1[7] = 0;
  i32x4 gz = {};
#if __clang_major__ >= 23
  i32x8 gz8 = {};
  __builtin_amdgcn_tensor_load_to_lds(g0, g1, gz, gz, gz8, 0);
#else
  __builtin_amdgcn_tensor_load_to_lds(g0, g1, gz, gz, 0);
#endif
}
#endif  // HAVE_TDM

// ---------------------------------------------------------------------------
// Weight transpose: dst[c][r] = src[r][c], src is R x C (row major).
// ---------------------------------------------------------------------------
__global__ __launch_bounds__(256)
void transpose_f32(const float* __restrict__ src, float* __restrict__ dst,
                   int R, int C) {
  __shared__ float t[32][33];
  const int tx = threadIdx.x & 31;
  const int ty = threadIdx.x >> 5;             // 0..7
  const int c0 = blockIdx.x * 32;
  const int r0 = blockIdx.y * 32;
#pragma unroll
  for (int i = ty; i < 32; i += 8)
    t[i][tx] = src[(size_t)(r0 + i) * C + (c0 + tx)];
  __syncthreads();
#pragma unroll
  for (int i = ty; i < 32; i += 8)
    dst[(size_t)(c0 + i) * R + (r0 + tx)] = t[tx][i];
}

// ---------------------------------------------------------------------------
// GEMM + bias: C[MxN] = A[MxK] * B[KxN] + bias[N], with B supplied
// pre-transposed as BT[NxK]. block = 128 (4 waves); BM=128, BN=128, BK=16;
// wave w -> rows [w*32, w*32+32) (2 m-tiles x 8 n-tiles = 16 accumulators).
// Both tiles are n/m-major k-minor, TDM-staged, double-buffered.
// ---------------------------------------------------------------------------
__global__ __launch_bounds__(128)
void gemm_bias_wmma(const float* __restrict__ A, const float* __restrict__ BT,
                    const float* __restrict__ bias, float* __restrict__ C,
                    int M, int N, int K) {
  __shared__ float As[2][128][18];   // 16 data + 2 pad (TDM pad codes 3,1)
  __shared__ float Bs[2][128][18];
  const int tid  = threadIdx.x;
  const int w    = tid >> 5;
  const int lane = tid & 31;
  const int half = lane >> 4;
  const int l15  = lane & 15;
  const int row0 = blockIdx.y * 128;
  const int col0 = blockIdx.x * 128;
  (void)M;

  v8f acc[2][8] = {};
  const int niter = K / 16;

#if HAVE_TDM
  if (tid < 32) {
    tdm_load_2d(A  + (size_t)row0 * K, (unsigned)(uintptr_t)&As[0][0][0],
                16, 128, (unsigned)K, 3, 1, 1);
    tdm_load_2d(BT + (size_t)col0 * K, (unsigned)(uintptr_t)&Bs[0][0][0],
                16, 128, (unsigned)K, 3, 1, 1);
  }
#endif

  for (int it = 0; it < niter; ++it) {
    const int cur = it & 1;
    const int k0 = it * 16;
#if HAVE_TDM
    if (it + 1 < niter) {
      if (tid < 32) {
        tdm_load_2d(A  + (size_t)row0 * K + (k0 + 16),
                    (unsigned)(uintptr_t)&As[cur ^ 1][0][0],
                    16, 128, (unsigned)K, 3, 1, 1);
        tdm_load_2d(BT + (size_t)col0 * K + (k0 + 16),
                    (unsigned)(uintptr_t)&Bs[cur ^ 1][0][0],
                    16, 128, (unsigned)K, 3, 1, 1);
      }
      __builtin_amdgcn_s_wait_tensorcnt(2);   // current tile (oldest 2) done
    } else {
      __builtin_amdgcn_s_wait_tensorcnt(0);
    }
#else
    for (int i = tid; i < 128 * 16; i += 128) {
      int r = i >> 4, c = i & 15;
      As[cur][r][c] = A[(size_t)(row0 + r) * K + (k0 + c)];
      Bs[cur][r][c] = BT[(size_t)(col0 + r) * K + (k0 + c)];
    }
#endif
    __syncthreads();

#pragma unroll
    for (int kk = 0; kk < 16; kk += 4) {
      v2f a[2];
#pragma unroll
      for (int mt = 0; mt < 2; ++mt) {
        a[mt].x = As[cur][w * 32 + mt * 16 + l15][kk + 2 * half];
        a[mt].y = As[cur][w * 32 + mt * 16 + l15][kk + 2 * half + 1];
      }
#pragma unroll
      for (int nt = 0; nt < 8; ++nt) {
        v2f b;  // B[k][n] = BT[n][k]
        b.x = Bs[cur][nt * 16 + l15][kk + 2 * half];
        b.y = Bs[cur][nt * 16 + l15][kk + 2 * half + 1];
        acc[0][nt] = wmma_k4(a[0], b, acc[0][nt]);
        acc[1][nt] = wmma_k4(a[1], b, acc[1][nt]);
      }
    }
    __syncthreads();   // readers done before next TDM overwrites spare buffer
  }

#pragma unroll
  for (int mt = 0; mt < 2; ++mt)
#pragma unroll
    for (int nt = 0; nt < 8; ++nt)
#pragma unroll
      for (int v = 0; v < 8; ++v) {
        int r = row0 + w * 32 + mt * 16 + v + 8 * half;
        int c = col0 + nt * 16 + l15;
        C[(size_t)r * N + c] = acc[mt][nt][v] + bias[c];
      }
}

// ---------------------------------------------------------------------------
// QK LayerNorm over head_dim + reshape. Q,K -> [B*NH, S, HD];
// V -> d-major [B*NH, HD, S] so attention V tiles are plain 2D TDM tiles.
// ---------------------------------------------------------------------------
__global__ __launch_bounds__(128)
void qk_ln_split(const float* __restrict__ qkv,
                 const float* __restrict__ qg, const float* __restrict__ qb,
                 const float* __restrict__ kg, const float* __restrict__ kb,
                 float* __restrict__ Qh, float* __restrict__ Kh,
                 float* __restrict__ VhT) {
  const int w    = threadIdx.x >> 5;
  const int lane = threadIdx.x & 31;
  const int row  = blockIdx.x * 4 + w;          // in [0, B*NH*S)
  const int s = row & (S_ - 1);
  const int h = (row / S_) & (NH_ - 1);
  const int b = row / (S_ * NH_);
  const size_t src  = (size_t)(b * S_ + s) * (3 * H_) + h * HD_;
  const size_t dst  = (size_t)((b * NH_ + h) * S_ + s) * HD_;
  const size_t vdst = (size_t)(b * NH_ + h) * HD_ * S_ + s;
  const int d0 = lane, d1 = lane + 32;

  float q0 = qkv[src + d0],          q1 = qkv[src + d1];
  float k0 = qkv[src + H_ + d0],     k1 = qkv[src + H_ + d1];
  float v0 = qkv[src + 2 * H_ + d0], v1 = qkv[src + 2 * H_ + d1];

  float qs = q0 + q1, qss = q0 * q0 + q1 * q1;
  float ks = k0 + k1, kss = k0 * k0 + k1 * k1;
#pragma unroll
  for (int off = 1; off < 32; off <<= 1) {
    qs  += __shfl_xor(qs,  off, 32);
    qss += __shfl_xor(qss, off, 32);
    ks  += __shfl_xor(ks,  off, 32);
    kss += __shfl_xor(kss, off, 32);
  }
  float qmu = qs * (1.f / HD_), qvar = qss * (1.f / HD_) - qmu * qmu;
  float kmu = ks * (1.f / HD_), kvar = kss * (1.f / HD_) - kmu * kmu;
  float qr = rsqrtf(qvar + EPS_), kr = rsqrtf(kvar + EPS_);

  Qh[dst + d0] = (q0 - qmu) * qr * qg[d0] + qb[d0];
  Qh[dst + d1] = (q1 - qmu) * qr * qg[d1] + qb[d1];
  Kh[dst + d0] = (k0 - kmu) * kr * kg[d0] + kb[d0];
  Kh[dst + d1] = (k1 - kmu) * kr * kg[d1] + kb[d1];
  VhT[vdst + (size_t)d0 * S_] = v0;             // transposed store
  VhT[vdst + (size_t)d1 * S_] = v1;
}

// ---------------------------------------------------------------------------
// Flash-style causal attention per (b, h, 64-row Q tile).
// block = 128 (4 waves); wave w owns q-rows [w*16, w*16+16).
// Q (once), K and V (double-buffered) all staged by TDM; V comes from the
// d-major layout so its tile is already [d][kcol].
// ---------------------------------------------------------------------------
__global__ __launch_bounds__(128)
void attn_wmma(const float* __restrict__ Qh, const float* __restrict__ Kh,
               const float* __restrict__ VhT, float* __restrict__ out) {
  __shared__ float Qs[64][66];        // 64 data + 2 pad (TDM codes 5,1)
  __shared__ float Ks[2][32][66];
  __shared__ float Vt[2][64][34];     // [d][kcol], 32 data + 2 pad (codes 4,1)
  __shared__ float Ps[4][16][34];     // per-wave P staging

  const int tid  = threadIdx.x;
  const int w    = tid >> 5;
  const int lane = tid & 31;
  const int half = lane >> 4;
  const int l15  = lane & 15;
  const int q0   = blockIdx.x * 64;
  const int h    = blockIdx.y;
  const int b    = blockIdx.z;
  const size_t base  = (size_t)(b * NH_ + h) * S_ * HD_;
  const size_t vbase = (size_t)(b * NH_ + h) * HD_ * S_;

  const int nkt = (q0 + 64) / 32;          // causal tile bound (>= 2)

#if HAVE_TDM
  if (tid < 32) {
    tdm_load_2d(Qh + base + (size_t)q0 * HD_, (unsigned)(uintptr_t)&Qs[0][0],
                64, 64, 64, 5, 1, 1);
    tdm_load_2d(Kh + base, (unsigned)(uintptr_t)&Ks[0][0][0],
                64, 32, 64, 5, 1, 1);
    tdm_load_2d(VhT + vbase, (unsigned)(uintptr_t)&Vt[0][0][0],
                32, 64, S_, 4, 1, 1);
  }
#else
  for (int i = tid; i < 64 * HD_; i += 128) {
    int r = i >> 6, d = i & 63;
    Qs[r][d] = Qh[base + (size_t)(q0 + r) * HD_ + d];
  }
#endif

  float mrow[8], lrow[8];
  v8f o[4] = {};
#pragma unroll
  for (int v = 0; v < 8; ++v) { mrow[v] = -3.0e38f; lrow[v] = 0.f; }
  const float scale = 0.125f;              // 64^-0.5

  for (int kt = 0; kt < nkt; ++kt) {
    const int cur = kt & 1;
    const int k0 = kt * 32;
#if HAVE_TDM
    if (kt + 1 < nkt) {
      if (tid < 32) {
        tdm_load_2d(Kh + base + (size_t)(k0 + 32) * HD_,
                    (unsigned)(uintptr_t)&Ks[cur ^ 1][0][0],
                    64, 32, 64, 5, 1, 1);
        tdm_load_2d(VhT + vbase + (size_t)(k0 + 32),
                    (unsigned)(uintptr_t)&Vt[cur ^ 1][0][0],
                    32, 64, S_, 4, 1, 1);
      }
      __builtin_amdgcn_s_wait_tensorcnt(2);
    } else {
      __builtin_amdgcn_s_wait_tensorcnt(0);
    }
#else
    for (int i = tid; i < 32 * HD_; i += 128) {
      int r = i >> 6, d = i & 63;
      Ks[cur][r][d] = Kh[base + (size_t)(k0 + r) * HD_ + d];
    }
    for (int i = tid; i < 64 * 32; i += 128) {
      int d = i >> 5, r = i & 31;
      Vt[cur][d][r] = VhT[vbase + (size_t)d * S_ + (k0 + r)];
    }
#endif
    __syncthreads();

    // ---- scores: S(16x32) = Q(16x64) * K^T(64x32) ----
    v8f sacc[2] = {};
#pragma unroll
    for (int kk = 0; kk < HD_; kk += 4) {
      v2f a;
      a.x = Qs[w * 16 + l15][kk + 2 * half];
      a.y = Qs[w * 16 + l15][kk + 2 * half + 1];
#pragma unroll
      for (int nt = 0; nt < 2; ++nt) {
        v2f bf;  // B[d][n] = K[n][d] = Ks[n][d]
        bf.x = Ks[cur][nt * 16 + l15][kk + 2 * half];
        bf.y = Ks[cur][nt * 16 + l15][kk + 2 * half + 1];
        sacc[nt] = wmma_k4(a, bf, sacc[nt]);
      }
    }

    // ---- scale + causal mask ----
#pragma unroll
    for (int nt = 0; nt < 2; ++nt) {
#pragma unroll
      for (int v = 0; v < 8; ++v) {
        int qrow = q0 + w * 16 + v + 8 * half;
        int kcol = k0 + nt * 16 + l15;
        float sv = sacc[nt][v] * scale;
        sacc[nt][v] = (kcol > qrow) ? -3.0e38f : sv;
      }
    }

    // ---- online softmax: row max over 16-lane groups ----
    float pmax[8];
#pragma unroll
    for (int v = 0; v < 8; ++v) pmax[v] = fmaxf(sacc[0][v], sacc[1][v]);
#pragma unroll
    for (int off = 1; off < 16; off <<= 1)
#pragma unroll
      for (int v = 0; v < 8; ++v)
        pmax[v] = fmaxf(pmax[v], __shfl_xor(pmax[v], off, 32));

    float sc[8];
#pragma unroll
    for (int v = 0; v < 8; ++v) {
      float mn = fmaxf(mrow[v], pmax[v]);
      sc[v] = __expf(mrow[v] - mn);
      mrow[v] = mn;
      lrow[v] *= sc[v];
    }
#pragma unroll
    for (int nt = 0; nt < 4; ++nt)
#pragma unroll
      for (int v = 0; v < 8; ++v) o[nt][v] *= sc[v];

    // ---- P = exp(S - m); row sums; stage P (k-contiguous) ----
    float psum[8] = {0, 0, 0, 0, 0, 0, 0, 0};
#pragma unroll
    for (int nt = 0; nt < 2; ++nt)
#pragma unroll
      for (int v = 0; v < 8; ++v) {
        float p = __expf(sacc[nt][v] - mrow[v]);
        psum[v] += p;
        Ps[w][v + 8 * half][nt * 16 + l15] = p;
      }
#pragma unroll
    for (int off = 1; off < 16; off <<= 1)
#pragma unroll
      for (int v = 0; v < 8; ++v) psum[v] += __shfl_xor(psum[v], off, 32);
#pragma unroll
    for (int v = 0; v < 8; ++v) lrow[v] += psum[v];

    // ---- O(16x64) += P(16x32) * V(32x64) ----
#pragma unroll
    for (int kk = 0; kk < 32; kk += 4) {
      v2f a;
      a.x = Ps[w][l15][kk + 2 * half];
      a.y = Ps[w][l15][kk + 2 * half + 1];
#pragma unroll
      for (int nt = 0; nt < 4; ++nt) {
        v2f bf;  // B[k][d] = V[k][d] = Vt[d][k]
        bf.x = Vt[cur][nt * 16 + l15][kk + 2 * half];
        bf.y = Vt[cur][nt * 16 + l15][kk + 2 * half + 1];
        o[nt] = wmma_k4(a, bf, o[nt]);
      }
    }
    __syncthreads();   // readers done before next TDM overwrites spare buffer
  }

  // ---- normalize + write back in [B,S,H] layout ----
#pragma unroll
  for (int nt = 0; nt < 4; ++nt)
#pragma unroll
    for (int v = 0; v < 8; ++v) {
      int qrow = q0 + w * 16 + v + 8 * half;
      int d = nt * 16 + l15;
      out[(size_t)(b * S_ + qrow) * H_ + h * HD_ + d] = o[nt][v] / lrow[v];
    }
}

// ---------------------------------------------------------------------------
extern "C" void kernel_launch(void* const* d_in, const int* in_sizes, int n_in,
                              void* d_out, int out_size, void* d_ws, size_t ws_size,
                              hipStream_t stream) {
  (void)in_sizes; (void)n_in; (void)out_size; (void)ws_size;
  const float* hidden = (const float*)d_in[0];
  const float* W_attn = (const float*)d_in[1];
  const float* b_attn = (const float*)d_in[2];
  const float* W_proj = (const float*)d_in[3];
  const float* b_proj = (const float*)d_in[4];
  const float* qg = (const float*)d_in[5];
  const float* qb = (const float*)d_in[6];
  const float* kg = (const float*)d_in[7];
  const float* kb = (const float*)d_in[8];

  float* ws   = (float*)d_ws;
  float* qkv  = ws;                                  // B*S*3H   = 12,582,912
  float* Qh   = qkv + (size_t)B_ * S_ * 3 * H_;      // B*NH*S*HD = 4,194,304
  float* Kh   = Qh  + (size_t)B_ * S_ * H_;
  float* VhT  = Kh  + (size_t)B_ * S_ * H_;          // d-major V
  float* WtA  = VhT + (size_t)B_ * S_ * H_;          // [3H][H] = 3,145,728
  float* WtP  = WtA + (size_t)3 * H_ * H_;           // [H][H]  = 1,048,576
  float* attn_out = ws;                              // reuse dead qkv region
  float* outp = (float*)d_out;

  // 0) one-time weight transposes (negligible: ~17 MB at 23.3 TB/s)
  transpose_f32<<<dim3((3 * H_) / 32, H_ / 32), 256, 0, stream>>>(
      W_attn, WtA, H_, 3 * H_);
  transpose_f32<<<dim3(H_ / 32, H_ / 32), 256, 0, stream>>>(
      W_proj, WtP, H_, H_);

  const dim3 blk(128);

  // 1) QKV projection: [4096x1024] x [1024x3072] + b_attn
  gemm_bias_wmma<<<dim3((3 * H_) / 128, (B_ * S_) / 128), blk, 0, stream>>>(
      hidden, WtA, b_attn, qkv, B_ * S_, 3 * H_, H_);

  // 2) QK LayerNorm + split into head-major Q/K and d-major V
  qk_ln_split<<<dim3((B_ * NH_ * S_) / 4), blk, 0, stream>>>(
      qkv, qg, qb, kg, kb, Qh, Kh, VhT);

  // 3) Causal flash attention
  attn_wmma<<<dim3(S_ / 64, NH_, B_), blk, 0, stream>>>(Qh, Kh, VhT, attn_out);

  // 4) Output projection: [4096x1024] x [1024x1024] + b_proj
  gemm_bias_wmma<<<dim3(H_ / 128, (B_ * S_) / 128), blk, 0, stream>>>(
      attn_out, WtP, b_proj, outp, B_ * S_, H_, H_);
}